// SLICSegmentation_5093831213735
// MI455X (gfx1250) — compile-verified
//
#include <hip/hip_runtime.h>
#include <math.h>
#include <limits.h>

// Problem constants (match reference)
#define H_ 224
#define W_ 224
#define HW_ (224 * 224)
#define KC_ 196          // clusters
#define B_ 8             // batch
#define NBH_ 10
#define S_ 16            // superpixel spacing: sqrt(HW/K)=16
#define MW_OVER_S 0.625f // M_WEIGHT / S = 10/16
#define KT_ 13           // cluster tiles (13*16 = 208 >= 196)

typedef float v2f __attribute__((ext_vector_type(2)));
typedef float v8f __attribute__((ext_vector_type(8)));

// ---------------------------------------------------------------------------
// Kernel 1: _find_minima. One block per batch; sequential scan over the 196
// grid centroids (occupancy is carried state -> must be serial). Block-wide
// min reductions over the <=20x20 window; claimed pixels kept as a flat-index
// list in LDS (<=196 entries, scanned only by the ~1 pixel matching mval).
// ---------------------------------------------------------------------------
__global__ void fm_kernel(const float* __restrict__ grad,
                          int* __restrict__ cy, int* __restrict__ cx,
                          float* __restrict__ cents_out) {
  const int b = blockIdx.x;
  const int t = threadIdx.x;
  const float* g = grad + (size_t)b * HW_;

  __shared__ float redf[256];
  __shared__ int   redi[256];
  __shared__ int   claimed[KC_];
  __shared__ int   nclaimed;

  if (t == 0) nclaimed = 0;
  __syncthreads();

  for (int k = 0; k < KC_; ++k) {
    // original grid point (window is always centered on cents0[k])
    const int cy0 = 16 * (k / 14) + 8;
    const int cx0 = 16 * (k % 14) + 8;
    const int y0 = max(cy0 - NBH_, 0), y1 = min(cy0 + NBH_, H_);
    const int x0 = max(cx0 - NBH_, 0), x1 = min(cx0 + NBH_, W_);
    const int ww = x1 - x0;
    const int n  = (y1 - y0) * ww;

    // pass 1: window min of g
    float mymin = 1e30f;
    for (int i = t; i < n; i += 256) {
      const int yy = y0 + i / ww;
      const int xx = x0 + i % ww;
      const float v = g[yy * W_ + xx];
      mymin = (v < mymin) ? v : mymin;
    }
    redf[t] = mymin;
    __syncthreads();
    for (int s = 128; s > 0; s >>= 1) {
      if (t < s) redf[t] = fminf(redf[t], redf[t + s]);
      __syncthreads();
    }
    const float mval = redf[0];

    // pass 2: smallest flat index with g==mval and not yet claimed
    int myidx = INT_MAX;
    const int nc = nclaimed;
    for (int i = t; i < n; i += 256) {
      const int yy = y0 + i / ww;
      const int xx = x0 + i % ww;
      const int flat = yy * W_ + xx;
      if (g[flat] == mval) {
        bool occ = false;
        for (int m = 0; m < nc; ++m) occ |= (claimed[m] == flat);
        if (!occ && flat < myidx) myidx = flat;
      }
    }
    redi[t] = myidx;
    __syncthreads();
    for (int s = 128; s > 0; s >>= 1) {
      if (t < s) redi[t] = min(redi[t], redi[t + s]);
      __syncthreads();
    }
    if (t == 0) {
      const int best = redi[0];
      int ny = cy0, nx = cx0;
      if (best != INT_MAX) {
        ny = best / W_;
        nx = best % W_;
        claimed[nclaimed++] = best;
      }
      const int idx = b * KC_ + k;
      cy[idx] = ny;
      cx[idx] = nx;
      cents_out[(size_t)idx * 2 + 0] = (float)ny;
      cents_out[(size_t)idx * 2 + 1] = (float)nx;
    }
    __syncthreads();
  }
}

// ---------------------------------------------------------------------------
// Kernel 2: gather initial cluster colors at centroid pixels.
// ---------------------------------------------------------------------------
__global__ void gather_colors_kernel(const float* __restrict__ x,
                                     const int* __restrict__ cy,
                                     const int* __restrict__ cx,
                                     float* __restrict__ colors) {
  const int i = blockIdx.x * blockDim.x + threadIdx.x;
  if (i >= B_ * KC_) return;
  const int b = i / KC_;
  const int yy = min(max(cy[i], 0), H_ - 1);
  const int xx = min(max(cx[i], 0), W_ - 1);
  const float* xp = x + (size_t)b * 3 * HW_;
  const int p = yy * W_ + xx;
  colors[i * 3 + 0] = xp[p];
  colors[i * 3 + 1] = xp[HW_ + p];
  colors[i * 3 + 2] = xp[2 * HW_ + p];
}

__global__ void zero_kernel(float* __restrict__ p, int n) {
  const int i = blockIdx.x * blockDim.x + threadIdx.x;
  if (i < n) p[i] = 0.0f;
}

// ---------------------------------------------------------------------------
// Kernel 3: SLIC assignment via V_WMMA_F32_16X16X4_F32.
// One wave32 owns 16 consecutive pixels (one row chunk; W%16==0).
//   A (16x4): row M = [c0, c1, c2, 1]        (lanes0-15: K0/K1, lanes16-31: K2/K3)
//   B (4x16): col N = [q0, q1, q2, -.5|q|^2] (VGPR r: K=r lo-half, K=2+r hi-half)
//   => D = cross - 0.5*|q|^2 ;  cd^2 = x2 - 2*D
// Cluster records are staged in LDS once per block (all 8 waves share a batch).
// A wave-uniform __any() test skips cluster tiles whose 32x32 windows cannot
// touch this row chunk -- EXEC stays all-ones, so the WMMA remains legal.
// Epilogue is branchless; a shfl_xor butterfly over the 16 cluster lanes per
// half-wave produces the first-min argmin.
// ---------------------------------------------------------------------------
__global__ void __launch_bounds__(256)
slic_assign_kernel(const float* __restrict__ x,
                   const int* __restrict__ cy,
                   const int* __restrict__ cx,
                   const float* __restrict__ colors,
                   float* __restrict__ labels,
                   float* __restrict__ acc,
                   int accumulate) {
  const int lane = threadIdx.x & 31;
  const int wave = (blockIdx.x * blockDim.x + threadIdx.x) >> 5;
  const int tilesPerBatch = HW_ / 16;              // 3136
  const int b     = wave / tilesPerBatch;          // uniform per block (392 blocks/batch)
  const int ptile = wave % tilesPerBatch;
  const int pbase = ptile * 16;
  const int yrow  = pbase / W_;                    // same row for all 16 pixels
  const int xb    = pbase % W_;

  const float* xp = x + (size_t)b * 3 * HW_;

  // ---- stage the 208 cluster records for this batch into LDS ----
  __shared__ int   s_cy[KT_ * 16];
  __shared__ int   s_cx[KT_ * 16];
  __shared__ float s_q[KT_ * 16][4];               // q0, q1, q2, -0.5*|q|^2
  {
    const int t = threadIdx.x;
    if (t < KT_ * 16) {
      const bool valid = (t < KC_);
      const int  kidx  = b * KC_ + (valid ? t : 0);
      const float q0 = valid ? colors[kidx * 3 + 0] : 0.0f;
      const float q1 = valid ? colors[kidx * 3 + 1] : 0.0f;
      const float q2 = valid ? colors[kidx * 3 + 2] : 0.0f;
      s_cy[t] = valid ? cy[kidx] : -100000;        // sentinel fails window tests
      s_cx[t] = valid ? cx[kidx] : -100000;
      s_q[t][0] = q0;
      s_q[t][1] = q1;
      s_q[t][2] = q2;
      s_q[t][3] = -0.5f * (q0 * q0 + q1 * q1 + q2 * q2);
    }
  }
  __syncthreads();

  // ---- build A and per-pixel x^2 ----
  const int  M0 = lane & 15;
  const bool hi = lane >= 16;
  const int  p0 = pbase + M0;
  const float c0 = xp[p0];
  const float c1 = xp[HW_ + p0];
  const float c2 = xp[2 * HW_ + p0];
  const float x2 = c0 * c0 + c1 * c1 + c2 * c2;

  v2f A;
  A.x = hi ? c2   : c0;   // K=2 | K=0
  A.y = hi ? 1.0f : c1;   // K=3 | K=1   (the 1.0 pairs with -0.5*|q|^2 in B)

  // broadcast x^2 of pixel row j (low half: M=j from lane j; high: M=8+j)
  float x2r[8];
#pragma unroll
  for (int j = 0; j < 8; ++j) {
    const float lo  = __shfl(x2, j, 32);
    const float hiv = __shfl(x2, 8 + j, 32);
    x2r[j] = hi ? hiv : lo;
  }

  float bd[8];
  int   bk[8];
#pragma unroll
  for (int j = 0; j < 8; ++j) { bd[j] = __builtin_inff(); bk[j] = -1; }

  const int N = lane & 15;
  for (int kt = 0; kt < KT_; ++kt) {
    const int sidx = kt * 16 + N;                  // this lane's cluster slot
    const int ycy  = s_cy[sidx];
    const int xcx  = s_cx[sidx];

    // can this cluster's 32x32 window touch this 16-pixel row chunk?
    const bool yok  = (yrow >= ycy - S_) && (yrow < ycy + S_);
    const bool xany = (xb + 15 >= xcx - S_) && (xb < xcx + S_);

    // wave-uniform skip: ballot-based, EXEC untouched (WMMA stays legal)
    if (!__any(yok && xany)) continue;

    v2f Bm;
    Bm.x = hi ? s_q[sidx][2] : s_q[sidx][0];       // K=2 | K=0
    Bm.y = hi ? s_q[sidx][3] : s_q[sidx][1];       // K=3 | K=1

    v8f Cz = {};
    v8f D = __builtin_amdgcn_wmma_f32_16x16x4_f32(
        false, A, false, Bm, (short)0, Cz, false, false);

    const float dyf = (float)(yrow - ycy);
    const float sp_y2 = dyf * dyf;
    const int k = kt * 16 + N;
#pragma unroll
    for (int j = 0; j < 8; ++j) {
      const int Mrow = (hi ? 8 : 0) + j;
      const int xpix = xb + Mrow;
      const bool ok  = yok && (xpix >= xcx - S_) && (xpix < xcx + S_);
      const float cd   = sqrtf(fmaxf(x2r[j] - 2.0f * D[j], 0.0f));
      const float dxf  = (float)(xpix - xcx);
      const float sp   = sqrtf(sp_y2 + dxf * dxf);
      const float dist = ok ? (cd + MW_OVER_S * sp) : __builtin_inff();
      const bool upd   = dist < bd[j];             // strict < keeps lowest k
      bd[j] = upd ? dist : bd[j];
      bk[j] = upd ? k    : bk[j];
    }
  }

  // butterfly argmin across the 16 cluster lanes of each half-wave
#pragma unroll
  for (int m = 1; m < 16; m <<= 1) {
#pragma unroll
    for (int j = 0; j < 8; ++j) {
      const float od = __shfl_xor(bd[j], m, 32);
      const int   ok = __shfl_xor(bk[j], m, 32);
      const bool  upd = (od < bd[j]) || (od == bd[j] && ok < bk[j]);
      bd[j] = upd ? od : bd[j];
      bk[j] = upd ? ok : bk[j];
    }
  }

  // lanes 0-7 handle M=0..7, lanes 16-23 handle M=8..15
  if ((lane & 15) < 8) {
    const int j    = lane & 7;
    const int Mrow = (hi ? 8 : 0) + j;
    const int p    = pbase + Mrow;
    const int lab  = bk[j];
    labels[(size_t)b * HW_ + p] = (float)lab;
    if (accumulate && lab >= 0) {
      float* a = acc + (size_t)(b * KC_ + lab) * 6;
      atomicAdd(a + 0, 1.0f);
      atomicAdd(a + 1, (float)yrow);
      atomicAdd(a + 2, (float)(xb + Mrow));
      atomicAdd(a + 3, xp[p]);
      atomicAdd(a + 4, xp[HW_ + p]);
      atomicAdd(a + 5, xp[2 * HW_ + p]);
    }
  }
}

// ---------------------------------------------------------------------------
// Kernel 4: centroid/color update (only where count > 0).
// jnp.round == round-half-to-even == rintf under default RNE mode.
// ---------------------------------------------------------------------------
__global__ void slic_update_kernel(const float* __restrict__ acc,
                                   int* __restrict__ cy, int* __restrict__ cx,
                                   float* __restrict__ colors) {
  const int i = blockIdx.x * blockDim.x + threadIdx.x;
  if (i >= B_ * KC_) return;
  const float cnt = acc[i * 6 + 0];
  if (cnt > 0.0f) {
    const float inv = 1.0f / cnt;
    const float fy = fminf(fmaxf(acc[i * 6 + 1] * inv, 0.0f), (float)(H_ - 1));
    const float fx = fminf(fmaxf(acc[i * 6 + 2] * inv, 0.0f), (float)(W_ - 1));
    cy[i] = (int)rintf(fy);
    cx[i] = (int)rintf(fx);
    colors[i * 3 + 0] = acc[i * 6 + 3] * inv;
    colors[i * 3 + 1] = acc[i * 6 + 4] * inv;
    colors[i * 3 + 2] = acc[i * 6 + 5] * inv;
  }
}

// ---------------------------------------------------------------------------
extern "C" void kernel_launch(void* const* d_in, const int* in_sizes, int n_in,
                              void* d_out, int out_size, void* d_ws, size_t ws_size,
                              hipStream_t stream) {
  const float* x    = (const float*)d_in[0];  // (8,3,224,224)
  const float* grad = (const float*)d_in[1];  // (8,1,224,224)

  float* out_cents  = (float*)d_out;                 // 8*196*2 floats (y,x)
  float* out_labels = out_cents + B_ * KC_ * 2;      // 8*224*224 floats

  // workspace layout
  int*   cyw    = (int*)d_ws;                        // 1568 ints
  int*   cxw    = cyw + B_ * KC_;                    // 1568 ints
  float* colors = (float*)(cxw + B_ * KC_);          // 1568*3 floats
  float* acc    = colors + B_ * KC_ * 3;             // 1568*6 floats

  // 1) find_minima: 8 blocks (one per batch), serial centroid scan inside
  fm_kernel<<<B_, 256, 0, stream>>>(grad, cyw, cxw, out_cents);

  // 2) initial cluster colors
  gather_colors_kernel<<<(B_ * KC_ + 255) / 256, 256, 0, stream>>>(x, cyw, cxw, colors);

  // 3) iter 1: zero accumulators, assign (+accumulate), update
  const int nacc = B_ * KC_ * 6;
  zero_kernel<<<(nacc + 255) / 256, 256, 0, stream>>>(acc, nacc);

  // 8 batches * 3136 pixel-tiles = 25088 waves; 8 waves/block -> 3136 blocks
  const int assign_blocks = (B_ * (HW_ / 16)) / 8;
  slic_assign_kernel<<<assign_blocks, 256, 0, stream>>>(
      x, cyw, cxw, colors, out_labels, acc, 1);

  slic_update_kernel<<<(B_ * KC_ + 255) / 256, 256, 0, stream>>>(acc, cyw, cxw, colors);

  // 4) iter 2: final labels only (post-update segment sums don't affect output)
  slic_assign_kernel<<<assign_blocks, 256, 0, stream>>>(
      x, cyw, cxw, colors, out_labels, acc, 0);
}